// octree_level_37228776522178
// MI455X (gfx1250) — compile-verified
//
#include <hip/hip_runtime.h>
#include <cstdint>

// ---------------------------------------------------------------------------
// Octree level build: dense-bitmap formulation.
//   Domain: leaf coords in [0,1024)^3  -> parents in [0,512)^3 = 2^27 slots.
//   Table:  1 byte (8-child mask) per parent slot = 128 MiB (fits MI455X L2).
//   Output: d_out = [N*3 floats parent_C (pad -1)] ++ [N*8 floats occupancy].
// ---------------------------------------------------------------------------

#define TABLE_WORDS  (1u << 25)   // 2^27 byte-slots packed 4-per-u32
#define CNT_THREADS  256
#define WORDS_PER_T  4            // each thread handles 4 words = 16 slots
#define CNT_BLOCKS   (TABLE_WORDS / (CNT_THREADS * WORDS_PER_T))  // 32768
#define SCAN_THREADS 1024
#define SCAN_CHUNK   (CNT_BLOCKS / SCAN_THREADS)                  // 32

// ---- gfx1250 async-copy helpers (inline asm; portable across toolchains) ----
__device__ __forceinline__ void async_b128_to_lds(uint32_t lds_addr, uint64_t gaddr) {
    asm volatile("global_load_async_to_lds_b128 %0, %1, off"
                 :: "v"(lds_addr), "v"(gaddr) : "memory");
}
__device__ __forceinline__ void wait_async0() {
    asm volatile("s_wait_asynccnt 0" ::: "memory");
}

__device__ __forceinline__ uint32_t nzbytes(uint32_t w) {
    uint32_t c = 0;
    c += (w & 0x000000FFu) ? 1u : 0u;
    c += (w & 0x0000FF00u) ? 1u : 0u;
    c += (w & 0x00FF0000u) ? 1u : 0u;
    c += (w & 0xFF000000u) ? 1u : 0u;
    return c;
}

// ---- K0: zero the mask table (uint4 stores, grid-stride) -------------------
__global__ void k_zero_table(uint4* __restrict__ t, int n4) {
    int i = blockIdx.x * blockDim.x + threadIdx.x;
    int stride = gridDim.x * blockDim.x;
    uint4 z; z.x = 0u; z.y = 0u; z.z = 0u; z.w = 0u;
    for (; i < n4; i += stride) t[i] = z;
}

// ---- K1: initialize outputs: parent_C region = -1, occupancy region = 0 ----
__global__ void k_init_out(float* __restrict__ out, long long n3, long long ntot) {
    long long i = (long long)blockIdx.x * blockDim.x + threadIdx.x;
    long long stride = (long long)gridDim.x * blockDim.x;
    for (; i < ntot; i += stride) out[i] = (i < n3) ? -1.0f : 0.0f;
}

// ---- K2: scatter leaves -> atomicOr child bit into parent's mask byte ------
__global__ void k_scatter(const float* __restrict__ leaf, uint32_t* __restrict__ table, int n) {
    int i = blockIdx.x * blockDim.x + threadIdx.x;
    if (i >= n) return;
    // prefetch the streaming input a few KB ahead (gfx1250 global_prefetch path)
    __builtin_prefetch(leaf + 3ll * i + 4096, 0, 1);
    int x = (int)floorf(leaf[3ll * i + 0]);
    int y = (int)floorf(leaf[3ll * i + 1]);
    int z = (int)floorf(leaf[3ll * i + 2]);
    uint32_t pk = ((uint32_t)(x >> 1) << 18) | ((uint32_t)(y >> 1) << 9) | (uint32_t)(z >> 1);
    uint32_t ci = (((uint32_t)x & 1u) << 2) | (((uint32_t)y & 1u) << 1) | ((uint32_t)z & 1u);
    uint32_t bit = (1u << ci) << (8u * (pk & 3u));
    atomicOr(&table[pk >> 2], bit);
}

// ---- K3: per-block count of occupied parent slots (async-to-LDS staging) ---
__global__ void k_count(const uint32_t* __restrict__ table, uint32_t* __restrict__ blockSums) {
    __shared__ __align__(16) uint32_t tile[CNT_THREADS * WORDS_PER_T];
    __shared__ uint32_t sred[CNT_THREADS];
    int tid = threadIdx.x;
    int blk = blockIdx.x;
    size_t wbase = (size_t)blk * (CNT_THREADS * WORDS_PER_T) + (size_t)tid * WORDS_PER_T;

    uint32_t lds_addr = (uint32_t)(uintptr_t)&tile[tid * WORDS_PER_T];
    uint64_t gaddr    = (uint64_t)(uintptr_t)(table + wbase);
    async_b128_to_lds(lds_addr, gaddr);
    wait_async0();
    __syncthreads();

    uint32_t c = 0;
    #pragma unroll
    for (int w = 0; w < WORDS_PER_T; ++w) c += nzbytes(tile[tid * WORDS_PER_T + w]);

    sred[tid] = c;
    __syncthreads();
    for (int s = CNT_THREADS / 2; s > 0; s >>= 1) {
        if (tid < s) sred[tid] += sred[tid + s];
        __syncthreads();
    }
    if (tid == 0) blockSums[blk] = sred[0];
}

// ---- K4: exclusive scan of 32768 block sums (single block, in place) -------
__global__ void k_scan(uint32_t* __restrict__ bs) {
    __shared__ uint32_t st[SCAN_THREADS];
    int tid = threadIdx.x;
    int base = tid * SCAN_CHUNK;

    uint32_t total = 0;
    #pragma unroll 4
    for (int k = 0; k < SCAN_CHUNK; ++k) total += bs[base + k];

    st[tid] = total;
    __syncthreads();
    for (int off = 1; off < SCAN_THREADS; off <<= 1) {
        uint32_t v = (tid >= off) ? st[tid - off] : 0u;
        __syncthreads();
        st[tid] += v;
        __syncthreads();
    }
    uint32_t run = st[tid] - total;   // exclusive prefix across threads
    #pragma unroll 4
    for (int k = 0; k < SCAN_CHUNK; ++k) {
        uint32_t v = bs[base + k];
        bs[base + k] = run;           // exclusive prefix per block
        run += v;
    }
}

// ---- K5: ordered compaction: emit parent rows + occupancy rows -------------
__global__ void k_compact(const uint32_t* __restrict__ table,
                          const uint32_t* __restrict__ blockOff,
                          float* __restrict__ outC,   // [N,3]
                          float* __restrict__ outOcc) // [N,8]
{
    __shared__ __align__(16) uint32_t tile[CNT_THREADS * WORDS_PER_T];
    __shared__ uint32_t sc[CNT_THREADS];
    int tid = threadIdx.x;
    int blk = blockIdx.x;
    size_t wbase = (size_t)blk * (CNT_THREADS * WORDS_PER_T) + (size_t)tid * WORDS_PER_T;

    uint32_t lds_addr = (uint32_t)(uintptr_t)&tile[tid * WORDS_PER_T];
    uint64_t gaddr    = (uint64_t)(uintptr_t)(table + wbase);
    async_b128_to_lds(lds_addr, gaddr);
    wait_async0();
    __syncthreads();

    uint32_t words[WORDS_PER_T];
    uint32_t c = 0;
    #pragma unroll
    for (int w = 0; w < WORDS_PER_T; ++w) {
        words[w] = tile[tid * WORDS_PER_T + w];
        c += nzbytes(words[w]);
    }

    // block-wide exclusive scan of per-thread counts
    sc[tid] = c;
    __syncthreads();
    for (int off = 1; off < CNT_THREADS; off <<= 1) {
        uint32_t v = (tid >= off) ? sc[tid - off] : 0u;
        __syncthreads();
        sc[tid] += v;
        __syncthreads();
    }
    uint32_t r = blockOff[blk] + (sc[tid] - c);

    #pragma unroll
    for (int w = 0; w < WORDS_PER_T; ++w) {
        uint32_t word = words[w];
        #pragma unroll
        for (int b = 0; b < 4; ++b) {
            uint32_t m = (word >> (8 * b)) & 0xFFu;
            if (m) {
                uint32_t pk = ((uint32_t)(wbase + w) << 2) | (uint32_t)b;
                float px = (float)((pk >> 18) & 511u);
                float py = (float)((pk >> 9) & 511u);
                float pz = (float)(pk & 511u);
                outC[3ull * r + 0] = px;
                outC[3ull * r + 1] = py;
                outC[3ull * r + 2] = pz;
                #pragma unroll
                for (int ci = 0; ci < 8; ++ci)
                    outOcc[8ull * r + ci] = (float)((m >> ci) & 1u);
                ++r;
            }
        }
    }
}

// ---------------------------------------------------------------------------
extern "C" void kernel_launch(void* const* d_in, const int* in_sizes, int n_in,
                              void* d_out, int out_size, void* d_ws, size_t ws_size,
                              hipStream_t stream) {
    (void)n_in; (void)out_size; (void)ws_size;
    const float* leaf = (const float*)d_in[0];
    const int N = in_sizes[0] / 3;

    float* out  = (float*)d_out;
    float* outC = out;                       // [N,3] parent coords (float)
    float* outO = out + 3ll * N;             // [N,8] occupancy

    uint32_t* table     = (uint32_t*)d_ws;           // 128 MiB mask table
    uint32_t* blockSums = table + TABLE_WORDS;       // 32768 u32

    // K0: zero table
    k_zero_table<<<8192, 256, 0, stream>>>((uint4*)table, (int)(TABLE_WORDS / 4));
    // K1: init outputs (-1 coords, 0 occupancy)
    long long n3 = 3ll * N, ntot = 11ll * N;
    k_init_out<<<4096, 256, 0, stream>>>(out, n3, ntot);
    // K2: scatter leaves into parent masks
    k_scatter<<<(N + 255) / 256, 256, 0, stream>>>(leaf, table, N);
    // K3: per-block occupied-slot counts
    k_count<<<CNT_BLOCKS, CNT_THREADS, 0, stream>>>(table, blockSums);
    // K4: exclusive scan of block counts
    k_scan<<<1, SCAN_THREADS, 0, stream>>>(blockSums);
    // K5: ordered compaction -> outputs
    k_compact<<<CNT_BLOCKS, CNT_THREADS, 0, stream>>>(table, blockSums, outC, outO);
}